// DTCWTFwd1_30253749633558
// MI455X (gfx1250) — compile-verified
//
#include <hip/hip_runtime.h>

// DTCWT forward level-1, fused single pass, register-blocked column filters.
// Input x: (4,32,512,512) f32; h0o: (5,) f32; h1o: (7,) f32.
// Output (flat concat): Yl (128*512*512), Yhr (128*6*256*256), Yhi (128*6*256*256).

#define IMG_H 512
#define IMG_W 512
#define NIMG  128              // B*C = 4*32
#define TH    64               // tile height
#define TW    64               // tile width
#define HALO  3                // max filter radius (7-tap)
#define EXT_H (TH + 2*HALO)    // 70
#define SW    72               // padded LDS row stride for s_in (floats); row base 16B-aligned
// stored column index = ext_col + 1 -> interior (ext col 3..66) at stored 4..67, 16B-aligned
#define NORI  6
#define OH    (IMG_H/2)
#define OW    (IMG_W/2)

typedef __attribute__((ext_vector_type(2))) float v2f;   // native vector for NT stores

__global__ __launch_bounds__(256)
void dtcwt_fwd_kernel(const float* __restrict__ x,
                      const float* __restrict__ h0,
                      const float* __restrict__ h1,
                      float* __restrict__ out)
{
    __shared__ float s_in[EXT_H][SW];      // 70x72 padded ext tile (20.2 KB)
    __shared__ float s_lo[EXT_H][TW];      // row-filtered, 5-tap h0 (17.9 KB)
    __shared__ float s_hi[EXT_H][TW];      // row-filtered, 7-tap h1 (17.9 KB)

    const int tid = threadIdx.x;
    const int gx0 = blockIdx.x * TW;       // tile origin (even, mult of 64)
    const int gy0 = blockIdx.y * TH;
    const int img = blockIdx.z;

    // Filter taps -> registers (uniform scalar loads).
    float f0[5], f1[7];
#pragma unroll
    for (int k = 0; k < 5; ++k) f0[k] = h0[k];
#pragma unroll
    for (int k = 0; k < 7; ++k) f1[k] = h1[k];

    const float* img_base = x + (size_t)img * (IMG_H * IMG_W);

    // ---------------- Phase 1: async global -> LDS tile stage ----------------
    // Symmetric mirror: i<0 -> -1-i ; i>=N -> 2N-1-i  (jnp.pad 'symmetric')
    {
        const unsigned lds_base = (unsigned)(size_t)(&s_in[0][0]);

        // 1a) Interior: 70 rows x 64 cols, b128 (4 floats/lane), fully aligned.
        //     item e: r = e>>4 (ext row), q = e&15 (16B quad within the 64 cols)
        for (int e = tid; e < EXT_H * 16; e += 256) {
            int r = e >> 4;
            int q = e & 15;
            int gy = gy0 + r - HALO;
            if (gy < 0) gy = -1 - gy; else if (gy >= IMG_H) gy = 2*IMG_H - 1 - gy;
            unsigned goff   = (unsigned)((gy * IMG_W + gx0) * 4 + q * 16);
            unsigned ldsoff = lds_base + (unsigned)(r * (SW*4) + 16 + q * 16);
            asm volatile("global_load_async_to_lds_b128 %0, %1, %2"
                         :: "v"(ldsoff), "v"(goff), "s"(img_base)
                         : "memory");
        }

        // 1b) Halo columns: 70 rows x 6 cols (ext cols 0,1,2 and 67,68,69), b32.
        for (int e = tid; e < EXT_H * 6; e += 256) {
            int r = e / 6;
            int i = e - r * 6;
            int cext = (i < 3) ? i : (64 + i);         // 0,1,2, 67,68,69
            int gy = gy0 + r - HALO;
            if (gy < 0) gy = -1 - gy; else if (gy >= IMG_H) gy = 2*IMG_H - 1 - gy;
            int gx = gx0 + cext - HALO;
            if (gx < 0) gx = -1 - gx; else if (gx >= IMG_W) gx = 2*IMG_W - 1 - gx;
            unsigned goff   = (unsigned)((gy * IMG_W + gx) * 4);
            unsigned ldsoff = lds_base + (unsigned)((r * SW + cext + 1) * 4);
            asm volatile("global_load_async_to_lds_b32 %0, %1, %2"
                         :: "v"(ldsoff), "v"(goff), "s"(img_base)
                         : "memory");
        }
        asm volatile("s_wait_asynccnt 0" ::: "memory");
    }
    __syncthreads();

    // ---------------- Phase 2: row filters (column-pair granularity) --------
    // Output col c (=2p): lo(c)=sum f0[k]*X[c+1+k], hi(c)=sum f1[k]*X[c+k]
    // Window X[c..c+7] is stored at float idx c+1..c+8 -> 5 aligned float2 reads.
    for (int e = tid; e < EXT_H * (TW/2); e += 256) {
        int r = e >> 5;            // / 32
        int p = e & 31;
        const float2* rowp = (const float2*)&s_in[r][0];
        float2 w0 = rowp[p],   w1 = rowp[p+1], w2 = rowp[p+2],
               w3 = rowp[p+3], w4 = rowp[p+4];
        float a0=w0.y, a1=w1.x, a2=w1.y, a3=w2.x, a4=w2.y, a5=w3.x, a6=w3.y, a7=w4.x;
        float lo0, lo1, hi0, hi1;
        lo0 = f0[0]*a1; lo0 = fmaf(f0[1],a2,lo0); lo0 = fmaf(f0[2],a3,lo0);
        lo0 = fmaf(f0[3],a4,lo0); lo0 = fmaf(f0[4],a5,lo0);
        lo1 = f0[0]*a2; lo1 = fmaf(f0[1],a3,lo1); lo1 = fmaf(f0[2],a4,lo1);
        lo1 = fmaf(f0[3],a5,lo1); lo1 = fmaf(f0[4],a6,lo1);
        hi0 = f1[0]*a0; hi0 = fmaf(f1[1],a1,hi0); hi0 = fmaf(f1[2],a2,hi0);
        hi0 = fmaf(f1[3],a3,hi0); hi0 = fmaf(f1[4],a4,hi0);
        hi0 = fmaf(f1[5],a5,hi0); hi0 = fmaf(f1[6],a6,hi0);
        hi1 = f1[0]*a1; hi1 = fmaf(f1[1],a2,hi1); hi1 = fmaf(f1[2],a3,hi1);
        hi1 = fmaf(f1[3],a4,hi1); hi1 = fmaf(f1[4],a5,hi1);
        hi1 = fmaf(f1[5],a6,hi1); hi1 = fmaf(f1[6],a7,hi1);
        float2 vlo; vlo.x = lo0; vlo.y = lo1;
        float2 vhi; vhi.x = hi0; vhi.y = hi1;
        ((float2*)&s_lo[r][0])[p] = vlo;
        ((float2*)&s_hi[r][0])[p] = vhi;
    }
    __syncthreads();

    // ---------------- Phase 3: column filters + q2c, register windows -------
    // thread = (strip s = tid/32, pair p = tid%32); wave == strip, lanes == pairs.
    // Strip covers output rows 8s..8s+7; needs ext rows 8s..8s+13 of Lo/Hi.
    {
        const int s = tid >> 5;
        const int p = tid & 31;

        float2 lo[14], hi[14];
#pragma unroll
        for (int i = 0; i < 14; ++i) {
            lo[i] = ((const float2*)&s_lo[8*s + i][0])[p];
            hi[i] = ((const float2*)&s_hi[8*s + i][0])[p];
        }

        float* Yl = out + (size_t)img * (IMG_H * IMG_W);
        const size_t YL_SZ  = (size_t)NIMG * IMG_H * IMG_W;
        const size_t SUB_SZ = (size_t)OH * OW;
        float* Yhr = out + YL_SZ;
        float* Yhi = out + YL_SZ + (size_t)NIMG * NORI * SUB_SZ;
        const float SQH = 0.70710678118654752440f;

#pragma unroll
        for (int j = 0; j < 8; j += 2) {           // one 2-row quad group per iter
            float lh[2][2], hl[2][2], hhv[2][2];
#pragma unroll
            for (int dy = 0; dy < 2; ++dy) {
                const int jj = j + dy;
                float llx=0.f, lly=0.f, lhx=0.f, lhy=0.f;
                float hlx=0.f, hly=0.f, hhx=0.f, hhy=0.f;
#pragma unroll
                for (int k = 0; k < 7; ++k) {      // 7-tap along H: LoHi, HiHi
                    float2 lv = lo[jj + k], hv = hi[jj + k];
                    lhx = fmaf(f1[k], lv.x, lhx);  lhy = fmaf(f1[k], lv.y, lhy);
                    hhx = fmaf(f1[k], hv.x, hhx);  hhy = fmaf(f1[k], hv.y, hhy);
                }
#pragma unroll
                for (int k = 0; k < 5; ++k) {      // 5-tap along H: LoLo, HiLo
                    float2 lv = lo[jj + 1 + k], hv = hi[jj + 1 + k];
                    llx = fmaf(f0[k], lv.x, llx);  lly = fmaf(f0[k], lv.y, lly);
                    hlx = fmaf(f0[k], hv.x, hlx);  hly = fmaf(f0[k], hv.y, hly);
                }
                lh[dy][0]=lhx; lh[dy][1]=lhy;
                hl[dy][0]=hlx; hl[dy][1]=hly;
                hhv[dy][0]=hhx; hhv[dy][1]=hhy;
                // Yl = LoLo, coalesced non-temporal 8B store (native vector type)
                v2f v; v.x = llx; v.y = lly;
                __builtin_nontemporal_store(
                    v, (v2f*)&Yl[(size_t)(gy0 + 8*s + jj) * IMG_W + (gx0 + 2*p)]);
            }
            // q2c (quad fully inside this thread): a=(0,0) b=(0,1) c=(1,0) d=(1,1)
            const int oy = (gy0 >> 1) + 4*s + (j >> 1);
            const int ox = (gx0 >> 1) + p;
            const size_t base = (size_t)img * NORI * SUB_SZ + (size_t)oy * OW + ox;

#define EMIT_SB(M, O_MD, O_PD)                                                     \
            {                                                                      \
                float a = M[0][0], b = M[0][1], c = M[1][0], d = M[1][1];           \
                __builtin_nontemporal_store((a - d) * SQH, &Yhr[base + (O_MD)*SUB_SZ]); \
                __builtin_nontemporal_store((b + c) * SQH, &Yhi[base + (O_MD)*SUB_SZ]); \
                __builtin_nontemporal_store((a + d) * SQH, &Yhr[base + (O_PD)*SUB_SZ]); \
                __builtin_nontemporal_store((b - c) * SQH, &Yhi[base + (O_PD)*SUB_SZ]); \
            }
            EMIT_SB(lh,  0, 5)   // LoHi -> 15deg, 165deg
            EMIT_SB(hhv, 1, 4)   // HiHi -> 45deg, 135deg
            EMIT_SB(hl,  2, 3)   // HiLo -> 75deg, 105deg
#undef EMIT_SB
        }
    }
}

extern "C" void kernel_launch(void* const* d_in, const int* in_sizes, int n_in,
                              void* d_out, int out_size, void* d_ws, size_t ws_size,
                              hipStream_t stream) {
    const float* x  = (const float*)d_in[0];   // (4,32,512,512)
    const float* h0 = (const float*)d_in[1];   // (5,)
    const float* h1 = (const float*)d_in[2];   // (7,)
    float* out = (float*)d_out;

    dim3 grid(IMG_W / TW, IMG_H / TH, NIMG);   // (8, 8, 128)
    dtcwt_fwd_kernel<<<grid, 256, 0, stream>>>(x, h0, h1, out);
}